// MonotonicMLP_72859825209962
// MI455X (gfx1250) — compile-verified
//
#include <hip/hip_runtime.h>

typedef __attribute__((ext_vector_type(2))) float v2f;
typedef __attribute__((ext_vector_type(8))) float v8f;

#define NPTS 1000
#define NSEG 999   // number of trapezoid segments

// ---------------------------------------------------------------------------
// Kernel 1: evaluate MLP on the 1000-point grid.
// One wave (block of 32) per 16-row tile; 63 tiles cover rows 0..1007 (rows
// >= 1000 computed but not stored). Layer 2 (1000x32 @ 32x32) is done with
// V_WMMA_F32_16X16X4_F32, accumulating K=32 in 8 steps, bias in C.
//
// Fragment layouts (CDNA5 ISA 7.12.2, 32-bit):
//   A 16x4 : lanes 0-15 -> M=lane, VGPR0=K0,VGPR1=K1 ; lanes 16-31 -> K2,K3
//   B 4x16 : lanes 0-15 -> N=lane, VGPR0=K0,VGPR1=K1 ; lanes 16-31 -> K2,K3
//   C 16x16: VGPR r, lanes 0-15 -> M=r, N=lane ; lanes 16-31 -> M=r+8
// ---------------------------------------------------------------------------
__global__ __launch_bounds__(32) void mlp_g_kernel(
    const float* __restrict__ w1, const float* __restrict__ b1,
    const float* __restrict__ w2, const float* __restrict__ b2,
    const float* __restrict__ w3, const float* __restrict__ b3,
    float* __restrict__ g_out)
{
    const int lane = threadIdx.x & 31;
    const int half = lane >> 4;     // 0: lanes 0-15, 1: lanes 16-31
    const int l    = lane & 15;
    const int row0 = blockIdx.x * 16;

    const int   myRow = row0 + l;                      // A-matrix row for this lane
    const float gv    = (float)myRow * (1.0f / 999.0f); // linspace(0,1,1000)

    // C init = bias b2 broadcast down rows (same for every VGPR/lane half)
    const float bias0 = b2[l];
    const float bias1 = b2[16 + l];
    v8f acc0, acc1;
#pragma unroll
    for (int r = 0; r < 8; ++r) { acc0[r] = bias0; acc1[r] = bias1; }

#pragma unroll
    for (int kk = 0; kk < 8; ++kk) {
        const int k0 = kk * 4 + half * 2;  // this lane's first K index of the step

        // A fragment: h1 = relu(grid*w1 + b1), computed on the fly
        v2f a;
        a[0] = fmaxf(fmaf(gv, w1[k0],     b1[k0]),     0.0f);
        a[1] = fmaxf(fmaf(gv, w1[k0 + 1], b1[k0 + 1]), 0.0f);

        // B fragments: w2 is (in,out) row-major, w2[k][n] = w2[k*32+n]
        v2f bv0, bv1;
        bv0[0] = w2[k0 * 32 + l];
        bv0[1] = w2[(k0 + 1) * 32 + l];
        bv1[0] = w2[k0 * 32 + 16 + l];
        bv1[1] = w2[(k0 + 1) * 32 + 16 + l];

        acc0 = __builtin_amdgcn_wmma_f32_16x16x4_f32(false, a, false, bv0,
                                                     (short)0, acc0, false, false);
        acc1 = __builtin_amdgcn_wmma_f32_16x16x4_f32(false, a, false, bv1,
                                                     (short)0, acc1, false, false);
    }

    // Layer 3: g = exp(relu(h2) @ w3 + b3) + eps.
    // Column for this lane: nTile*16 + l. Row M = r + 8*half lives across the
    // 16 lanes of this half -> butterfly-reduce within the half.
    const float w3a = w3[l];
    const float w3b = w3[16 + l];
    float p[8];
#pragma unroll
    for (int r = 0; r < 8; ++r)
        p[r] = fmaxf(acc0[r], 0.0f) * w3a + fmaxf(acc1[r], 0.0f) * w3b;

#pragma unroll
    for (int m = 1; m < 16; m <<= 1) {
#pragma unroll
        for (int r = 0; r < 8; ++r)
            p[r] += __shfl_xor(p[r], m, 32);
    }

    if (l == 0) {
        const float b3v = b3[0];
#pragma unroll
        for (int r = 0; r < 8; ++r) {
            const int row = row0 + half * 8 + r;
            if (row < NPTS) g_out[row] = expf(p[r] + b3v) + 1e-6f;
        }
    }
}

// ---------------------------------------------------------------------------
// Kernel 2: trapezoid cumsum + normalization -> cum[0..999] in [0,1].
// Single block of 1024 threads, Hillis-Steele inclusive scan in LDS.
// cum[i] = (sum_{t<i} (g[t]+g[t+1]))*dx/2 / total, cum[0]=0, cum[999]=1.
// ---------------------------------------------------------------------------
__global__ __launch_bounds__(1024) void cdf_scan_kernel(
    const float* __restrict__ g, float* __restrict__ cum)
{
    __shared__ float buf[1024];
    const int t = threadIdx.x;

    float v = 0.0f;
    if (t < NSEG) v = (g[t] + g[t + 1]) * (0.5f / 999.0f);
    buf[t] = v;
    __syncthreads();

#pragma unroll
    for (int off = 1; off < 1024; off <<= 1) {
        const float add = (t >= off) ? buf[t - off] : 0.0f;
        __syncthreads();
        buf[t] += add;
        __syncthreads();
    }

    if (t < NPTS) {
        const float total = buf[NSEG - 1];
        const float c = (t == 0) ? 0.0f : buf[t - 1];
        cum[t] = c / total;
    }
}

// ---------------------------------------------------------------------------
// Kernel 3: memory-bound piecewise-linear interpolation (the 22us roofline).
// CDF table (4 KB) staged in LDS; 128-bit loads/stores via float4.
// ---------------------------------------------------------------------------
__device__ __forceinline__ float interp_one(float xv, const float* lut) {
    const float s = xv * 999.0f;
    int idx = (int)s;                       // floor for s >= 0
    idx = idx < 0 ? 0 : (idx > 998 ? 998 : idx);
    const float alpha = s - (float)idx;
    const float lo = lut[idx];
    const float hi = lut[idx + 1];
    return fmaf(alpha, hi - lo, lo);
}

__global__ __launch_bounds__(256) void interp_kernel(
    const float* __restrict__ x, const float* __restrict__ cum,
    float* __restrict__ out, long long n4, long long n)
{
    __shared__ float lut[NPTS];
    for (int i = threadIdx.x; i < NPTS; i += blockDim.x) lut[i] = cum[i];
    __syncthreads();

    const long long tid    = (long long)blockIdx.x * blockDim.x + threadIdx.x;
    const long long stride = (long long)gridDim.x * blockDim.x;

    const float4* __restrict__ x4 = (const float4*)x;
    float4* __restrict__ o4       = (float4*)out;

    for (long long i = tid; i < n4; i += stride) {
        const float4 xv = x4[i];
        float4 ov;
        ov.x = interp_one(xv.x, lut);
        ov.y = interp_one(xv.y, lut);
        ov.z = interp_one(xv.z, lut);
        ov.w = interp_one(xv.w, lut);
        o4[i] = ov;
    }
    // tail (n not divisible by 4 — not hit for this shape, kept for safety)
    for (long long i = 4 * n4 + tid; i < n; i += stride)
        out[i] = interp_one(x[i], lut);
}

// ---------------------------------------------------------------------------
extern "C" void kernel_launch(void* const* d_in, const int* in_sizes, int n_in,
                              void* d_out, int out_size, void* d_ws, size_t ws_size,
                              hipStream_t stream)
{
    (void)n_in; (void)out_size; (void)ws_size;

    const float* x  = (const float*)d_in[0];
    const float* w1 = (const float*)d_in[1];
    const float* b1 = (const float*)d_in[2];
    const float* w2 = (const float*)d_in[3];
    const float* b2 = (const float*)d_in[4];
    const float* w3 = (const float*)d_in[5];
    const float* b3 = (const float*)d_in[6];
    float* out = (float*)d_out;

    float* g   = (float*)d_ws;   // 1000 floats
    float* cum = g + 1024;       // 1000 floats (offset 4 KB)

    const long long n = (long long)in_sizes[0];

    // Phase 1: MLP integrand on the grid (63 tiles of 16 rows, 1 wave each)
    mlp_g_kernel<<<63, 32, 0, stream>>>(w1, b1, w2, b2, w3, b3, g);

    // Phase 2: normalized cumulative trapezoid CDF
    cdf_scan_kernel<<<1, 1024, 0, stream>>>(g, cum);

    // Phase 3: 67M-element table interpolation (HBM-bound)
    const long long n4 = n >> 2;
    int blocks = (int)((n4 + 255) / 256);
    if (blocks > 8192) blocks = 8192;
    if (blocks < 1) blocks = 1;
    interp_kernel<<<blocks, 256, 0, stream>>>(x, cum, out, n4, n);
}